// GCNEncoder_51702816309674
// MI455X (gfx1250) — compile-verified
//
#include <hip/hip_runtime.h>
#include <stdint.h>

typedef __attribute__((ext_vector_type(2))) float v2f;
typedef __attribute__((ext_vector_type(8))) float v8f;

// ---------------------------------------------------------------------------
// Preprocessing kernels
// ---------------------------------------------------------------------------

// xpad[N][16]: copy x[N][15], zero column 15 (pads K to a multiple of 4 for WMMA)
__global__ void padx_kernel(const float* __restrict__ x, float* __restrict__ xpad, int N) {
    int i = blockIdx.x * blockDim.x + threadIdx.x;
    if (i < N * 16) {
        int n = i >> 4, c = i & 15;
        xpad[i] = (c < 15) ? x[n * 15 + c] : 0.0f;
    }
}

// deg[d] += w[e]  over real edges (self-loop +1 added in dinv_kernel)
__global__ void degree_kernel(const int* __restrict__ dst, const float* __restrict__ w,
                              float* __restrict__ deg, int E) {
    int e = blockIdx.x * blockDim.x + threadIdx.x;
    if (e < E) atomicAdd(&deg[dst[e]], w[e]);
}

// deg -> dinv = rsqrt(deg + 1)   (in place)
__global__ void dinv_kernel(float* __restrict__ deg, int N) {
    int i = blockIdx.x * blockDim.x + threadIdx.x;
    if (i < N) {
        float d = deg[i] + 1.0f;  // self-loop weight 1.0
        deg[i] = (d > 0.0f) ? rsqrtf(d) : 0.0f;
    }
}

// norm[e] = dinv[s]*w*dinv[d];  selfnorm[i] = dinv[i]^2
__global__ void norm_kernel(const int* __restrict__ src, const int* __restrict__ dst,
                            const float* __restrict__ w, const float* __restrict__ dinv,
                            float* __restrict__ norm, float* __restrict__ selfn,
                            int E, int N) {
    int i = blockIdx.x * blockDim.x + threadIdx.x;
    if (i < E) norm[i] = dinv[src[i]] * w[i] * dinv[dst[i]];
    if (i < N) { float v = dinv[i]; selfn[i] = v * v; }
}

// ---------------------------------------------------------------------------
// Dense transform: hw[N][64] = h[N][K] @ W[Kw][64]   via V_WMMA_F32_16X16X4_F32
//   Block = 128 threads (4 waves); each wave owns one 16-row chunk, all 64 cols.
//   A frag (16x4 f32): lane L: row = lm, k = k0 + 2*(L>>4) .. +1   (2 VGPRs)
//   B frag (4x16 f32): lane L: col = lm, k = k0 + 2*(L>>4) .. +1   (2 VGPRs)
//   C/D (16x16 f32):   lane L: col = lm, VGPR j -> row j + 8*(L>>4)
//   LDS W layout is PAIR-INTERLEAVED: lds[kp*128 + c*2 + j] = W[2*kp + j][c]
//   so every B fragment is one contiguous 8B ds_load_b64 (no repacking movs).
// ---------------------------------------------------------------------------
__global__ void gemm_wmma_kernel(const float* __restrict__ h, const float* __restrict__ W,
                                 float* __restrict__ hw, int N, int K, int Kw) {
    __shared__ float lds_w[64 * 64];
    // Stage W into LDS, interleaving consecutive k-rows (rows >= Kw zero padded)
    for (int i = threadIdx.x; i < K * 64; i += blockDim.x) {
        int j  = i & 1;
        int c  = (i >> 1) & 63;
        int kp = i >> 7;
        int r  = kp * 2 + j;
        lds_w[i] = (r < Kw) ? W[r * 64 + c] : 0.0f;
    }
    __syncthreads();

    const int lane = threadIdx.x & 31;
    const int wave = threadIdx.x >> 5;
    const int nChunks = (N + 15) >> 4;
    const int chunk = blockIdx.x * 4 + wave;   // wave-uniform
    if (chunk >= nChunks) return;              // whole wave exits: EXEC stays all-1s

    const int lm   = lane & 15;
    const int half = lane >> 4;                // 0 or 1
    const int row0 = chunk << 4;
    int arow = row0 + lm; if (arow >= N) arow = N - 1;
    const size_t abase = (size_t)arow * K;
    const int koff = half * 2;

    v8f acc0 = {}, acc1 = {}, acc2 = {}, acc3 = {};
    for (int k0 = 0; k0 < K; k0 += 4) {
        const int kb = k0 + koff;              // even -> 8B aligned
        v2f a = *(const v2f*)(h + abase + kb);
        // pair index for this lane-half's two k values
        const float* wb = lds_w + (size_t)(kb >> 1) * 128 + lm * 2;
        v2f b0 = *(const v2f*)(wb + 0);
        v2f b1 = *(const v2f*)(wb + 32);
        v2f b2 = *(const v2f*)(wb + 64);
        v2f b3 = *(const v2f*)(wb + 96);
        acc0 = __builtin_amdgcn_wmma_f32_16x16x4_f32(false, a, false, b0, (short)0, acc0, false, false);
        acc1 = __builtin_amdgcn_wmma_f32_16x16x4_f32(false, a, false, b1, (short)0, acc1, false, false);
        acc2 = __builtin_amdgcn_wmma_f32_16x16x4_f32(false, a, false, b2, (short)0, acc2, false, false);
        acc3 = __builtin_amdgcn_wmma_f32_16x16x4_f32(false, a, false, b3, (short)0, acc3, false, false);
    }

    #pragma unroll
    for (int j = 0; j < 8; ++j) {
        int r = row0 + half * 8 + j;
        if (r < N) {
            float* orow = hw + (size_t)r * 64 + lm;
            orow[0]  = acc0[j];
            orow[16] = acc1[j];
            orow[32] = acc2[j];
            orow[48] = acc3[j];
        }
    }
}

// out[n][f] = b[f]  (bias pre-load so scatter adds on top; matches segsum + b)
__global__ void init_bias_kernel(const float* __restrict__ b, float* __restrict__ out,
                                 long long total) {
    long long i = (long long)blockIdx.x * blockDim.x + threadIdx.x;
    if (i < total) out[i] = b[i & 63];
}

// Scatter-add: for each augmented edge (E real + N self loops), 64 feature lanes:
//   out[d][f] += norm * hw[s][f]
__global__ void aggregate_kernel(const int* __restrict__ src, const int* __restrict__ dst,
                                 const float* __restrict__ norm, const float* __restrict__ selfn,
                                 const float* __restrict__ hw, float* __restrict__ out,
                                 int E, int N) {
    __shared__ int   ss[4], sd[4];
    __shared__ float sn[4];
    const long long ebase = (long long)blockIdx.x * 4;
    const long long total = (long long)E + N;
    if (threadIdx.x < 4) {
        long long idx = ebase + threadIdx.x;
        if (idx < total) {
            if (idx < E) {
                ss[threadIdx.x] = src[idx];
                sd[threadIdx.x] = dst[idx];
                sn[threadIdx.x] = norm[idx];
            } else {
                int i = (int)(idx - E);
                ss[threadIdx.x] = i;
                sd[threadIdx.x] = i;
                sn[threadIdx.x] = selfn[i];
            }
        }
    }
    __syncthreads();
    const int el = threadIdx.x >> 6;          // local edge 0..3
    const int f  = threadIdx.x & 63;          // feature lane
    if (ebase + el < total) {
        const float v = sn[el] * hw[(size_t)ss[el] * 64 + f];
        atomicAdd(&out[(size_t)sd[el] * 64 + f], v);
    }
}

// mode 0: ReLU in place; mode 1: sigmoid in place
__global__ void act_kernel(float* __restrict__ h, long long total, int mode) {
    long long i = (long long)blockIdx.x * blockDim.x + threadIdx.x;
    if (i < total) {
        float v = h[i];
        h[i] = (mode == 0) ? fmaxf(v, 0.0f) : 1.0f / (1.0f + expf(-v));
    }
}

// ---------------------------------------------------------------------------
// Host launcher
// ---------------------------------------------------------------------------
static inline int cdiv_ll(long long a, long long b) { return (int)((a + b - 1) / b); }

extern "C" void kernel_launch(void* const* d_in, const int* in_sizes, int n_in,
                              void* d_out, int out_size, void* d_ws, size_t ws_size,
                              hipStream_t stream) {
    const float* x   = (const float*)d_in[0];
    const int*   ei  = (const int*)  d_in[1];   // [2][E]
    const float* ew  = (const float*)d_in[2];
    const float* W0  = (const float*)d_in[3];   // [15][64]
    const float* b0  = (const float*)d_in[4];
    const float* Wm  = (const float*)d_in[5];   // [6][64][64]
    const float* bm  = (const float*)d_in[6];   // [6][64]
    const float* Wl  = (const float*)d_in[7];   // [64][64]
    const float* bl  = (const float*)d_in[8];

    const int N = in_sizes[0] / 15;
    const int E = in_sizes[2];
    const int* src = ei;
    const int* dst = ei + E;

    // Workspace layout (floats)
    float* ws    = (float*)d_ws;
    float* xpad  = ws;                          // N*16
    float* dinv  = xpad  + (size_t)N * 16;      // N (degree -> dinv in place)
    float* normv = dinv  + N;                   // E
    float* selfn = normv + E;                   // N
    float* h1    = selfn + N;                   // N*64  (GEMM output)
    float* hA    = h1    + (size_t)N * 64;      // N*64  (activations ping)
    float* hB    = hA    + (size_t)N * 64;      // N*64  (activations pong)

    const long long NF  = (long long)N * 64;
    const long long EN  = (long long)E + N;
    const int nChunks   = (N + 15) >> 4;

    // --- GCN normalization (once) ---
    hipMemsetAsync(dinv, 0, (size_t)N * sizeof(float), stream);
    padx_kernel  <<<cdiv_ll((long long)N * 16, 256), 256, 0, stream>>>(x, xpad, N);
    degree_kernel<<<cdiv_ll(E, 256), 256, 0, stream>>>(dst, ew, dinv, E);
    dinv_kernel  <<<cdiv_ll(N, 256), 256, 0, stream>>>(dinv, N);
    norm_kernel  <<<cdiv_ll(E > N ? E : N, 256), 256, 0, stream>>>(src, dst, ew, dinv,
                                                                   normv, selfn, E, N);

    // --- one GCN layer: gemm -> bias init -> scatter-add -> activation ---
    auto run_layer = [&](const float* hin, int K, int Kw, const float* W, const float* b,
                         float* out, int actmode) {
        gemm_wmma_kernel<<<cdiv_ll(nChunks, 4), 128, 0, stream>>>(hin, W, h1, N, K, Kw);
        init_bias_kernel<<<cdiv_ll(NF, 256), 256, 0, stream>>>(b, out, NF);
        aggregate_kernel<<<cdiv_ll(EN * 64, 256), 256, 0, stream>>>(src, dst, normv, selfn,
                                                                    h1, out, E, N);
        act_kernel<<<cdiv_ll(NF, 256), 256, 0, stream>>>(out, NF, actmode);
    };

    // Layer 0: [N,16(pad)] @ [16(pad),64], ReLU
    run_layer(xpad, 16, 15, W0, b0, hA, /*relu*/0);

    // Middle layers 1..6
    const float* hin = hA;
    float*       ho  = hB;
    for (int i = 0; i < 6; ++i) {
        run_layer(hin, 64, 64, Wm + (size_t)i * 64 * 64, bm + (size_t)i * 64, ho, /*relu*/0);
        const float* t = hin; hin = ho; ho = (float*)t;
    }

    // Final layer + sigmoid, directly into d_out
    run_layer(hin, 64, 64, Wl, bl, (float*)d_out, /*sigmoid*/1);
}